// EfficientPairEmbed_82617990906537
// MI455X (gfx1250) — compile-verified
//
#include <hip/hip_runtime.h>
#include <cstdint>

// EfficientPairEmbed on MI455X (gfx1250):
//  - async global->LDS staging of gathered embedding rows (ASYNCcnt path)
//  - f32 WMMA 16x16x4 for the per-edge [8x50]x[50] einsum (diagonal of a
//    16-edge x 16-edge product; f32 matrix path keeps reference precision)

#define NUM_G     50
#define NUM_H     8
#define NUM_EL    100
#define RBF_R     12.0f
#define TILE      16
#define ROW_F     416   // padded LDS row stride in floats (1664 B, 8B aligned)

typedef float v2f __attribute__((ext_vector_type(2)));
typedef float v8f __attribute__((ext_vector_type(8)));

__global__ __launch_bounds__(64) void EfficientPairEmbed_kernel(
    const int*   __restrict__ anum,
    const int*   __restrict__ edge_index,
    const float* __restrict__ dist,
    const float* __restrict__ embedding,
    float*       __restrict__ out,
    int nEdges)
{
  __shared__ __align__(16) float rows[2][TILE][ROW_F];
  __shared__ __align__(16) float outD[2][TILE * NUM_H];

  const int lane = threadIdx.x & 31;
  const int wave = threadIdx.x >> 5;
  const int tile = blockIdx.x * 2 + wave;
  if (tile * TILE >= nEdges) return;

  const int t = lane & 15;                      // edge slot within tile
  int e = tile * TILE + t;
  if (e >= nEdges) e = nEdges - 1;              // safety clamp (E%16==0 anyway)

  // Per-lane edge info (upper/lower lane halves duplicate slot t).
  const int   sN = edge_index[e];
  const int   dN = edge_index[nEdges + e];
  const uint32_t rowOffB =
      (uint32_t)(anum[sN] * NUM_EL + anum[dN]) * (uint32_t)(NUM_H * NUM_G * 4);
  const float dval = dist[e];

  // ---- zero the per-row LDS pad (floats 400..415) so pad*0 never sees NaN
  #pragma unroll
  for (int j = 0; j < 8; ++j) {
    int idx = j * 32 + lane;                    // 0..255
    rows[wave][idx >> 4][400 + (idx & 15)] = 0.0f;
  }

  // ---- async-stage 16 embedding rows (1600 B each) into LDS --------------
  const uint32_t ldsRowsBase = (uint32_t)(uintptr_t)&rows[wave][0][0];
  const uint64_t gbase       = (uint64_t)(uintptr_t)embedding;
  #pragma unroll
  for (int i = 0; i < 50; ++i) {
    int u   = i * 32 + lane;                    // 0..1599 : 16B chunk id
    int row = u / 100;                          // which of the 16 rows
    int rem = u - row * 100;                    // 16B chunk within row
    uint32_t ro    = (uint32_t)__shfl((int)rowOffB, row, 32);
    uint32_t voff  = ro + (uint32_t)rem * 16u;  // byte offset vs embedding base
    uint32_t laddr = ldsRowsBase + (uint32_t)row * (ROW_F * 4u)
                                 + (uint32_t)rem * 16u;
    asm volatile("global_load_async_to_lds_b128 %0, %1, %2"
                 :: "v"(laddr), "v"(voff), "s"(gbase) : "memory");
  }

  // ---- RBF A-operand directly in WMMA 16x4 f32 layout --------------------
  // lane L holds A[t, k] for k = {dlt, dlt+1} of each K=4 step (dlt = 2*(L>=16))
  const float stddev = RBF_R / (float)NUM_G;
  const float coeff  = -0.5f / (stddev * stddev);
  const float step   = RBF_R / (float)(NUM_G - 1);
  const int   dlt    = (lane >> 4) * 2;
  v2f a[13];
  #pragma unroll
  for (int k = 0; k < 13; ++k) {
    int   g0 = 4 * k + dlt;
    float f0 = dval - (float)g0 * step;
    float f1 = dval - (float)(g0 + 1) * step;
    a[k].x = (g0     < NUM_G) ? __expf(coeff * f0 * f0) : 0.0f;
    a[k].y = (g0 + 1 < NUM_G) ? __expf(coeff * f1 * f1) : 0.0f;
  }

  // ---- wait for async rows, then WMMA einsum -----------------------------
  asm volatile("s_wait_asynccnt 0" ::: "memory");
  __builtin_amdgcn_wave_barrier();

  v8f c[NUM_H];
  const v8f cz = {0.f, 0.f, 0.f, 0.f, 0.f, 0.f, 0.f, 0.f};
  #pragma unroll
  for (int h = 0; h < NUM_H; ++h) c[h] = cz;

  // B[k, t'] for this lane: rows[wave][t][h*50 + 4k + dlt + {0,1}]
  const float* bBase =
      (const float*)__builtin_assume_aligned(&rows[wave][0][0], 16);
  const int bLane = t * ROW_F + dlt;            // 8B-aligned float offset

  #pragma unroll
  for (int k = 0; k < 13; ++k) {
    const v2f av = a[k];
    #pragma unroll
    for (int h = 0; h < NUM_H; ++h) {
      const v2f bv = *(const v2f*)(bBase + bLane + h * NUM_G + 4 * k);
      c[h] = __builtin_amdgcn_wmma_f32_16x16x4_f32(
                 /*neg_a=*/false, av, /*neg_b=*/false, bv,
                 /*c_mod=*/(short)0, c[h], /*reuse_a=*/false, /*reuse_b=*/false);
    }
  }

  // ---- extract diagonal D[t,t] -> outD[t*8 + h] --------------------------
  // C/D layout: VGPR r, lanes 0-15 => M=r, lanes 16-31 => M=8+r; N = lane&15.
  // Diagonal of VGPR r lives in lane r (t=r) and lane r+24 (t=8+r).
  #pragma unroll
  for (int r = 0; r < 8; ++r) {
    if (lane == r || lane == r + 24) {
      const int tOut = ((lane >> 4) << 3) + r;  // r or 8+r
      #pragma unroll
      for (int h = 0; h < NUM_H; ++h)
        outD[wave][tOut * NUM_H + h] = c[h][r];
    }
  }
  __builtin_amdgcn_wave_barrier();
  asm volatile("" ::: "memory");

  // ---- coalesced store: 128 floats per tile ------------------------------
  const float4 o4 = *(const float4*)&outD[wave][lane * 4];
  ((float4*)out)[(size_t)tile * 32 + lane] = o4;
}

extern "C" void kernel_launch(void* const* d_in, const int* in_sizes, int n_in,
                              void* d_out, int out_size, void* d_ws, size_t ws_size,
                              hipStream_t stream) {
  (void)n_in; (void)out_size; (void)d_ws; (void)ws_size;
  const int*   anum       = (const int*)d_in[0];
  const int*   edge_index = (const int*)d_in[1];
  const float* dist       = (const float*)d_in[2];
  const float* embedding  = (const float*)d_in[3];
  float*       out        = (float*)d_out;

  const int nEdges = in_sizes[2];               // dist has E elements
  const int tiles  = (nEdges + TILE - 1) / TILE;
  const int blocks = (tiles + 1) / 2;           // 2 waves (tiles) per block

  EfficientPairEmbed_kernel<<<blocks, 64, 0, stream>>>(
      anum, edge_index, dist, embedding, out, nEdges);
}